// LengthRegulator_24790551233198
// MI455X (gfx1250) — compile-verified
//
#include <hip/hip_runtime.h>
#include <stdint.h>
#include <stddef.h>

#ifndef __has_builtin
#define __has_builtin(x) 0
#endif

#define NB  32
#define NTK 512          // tokens per batch
#define ND  512          // feature dim
#define ND4 (ND / 4)     // float4s per row (128)
#define NWAVES (NTK / 32)

// 16-byte vector types
typedef float vfloat4 __attribute__((ext_vector_type(4)));
typedef int   v4i     __attribute__((vector_size(4 * sizeof(int))));
typedef __attribute__((address_space(3))) v4i v4i_lds;

// ---- gfx1250 async LDS-DMA path (compile-probed via __has_builtin) --------
#if defined(__HIP_DEVICE_COMPILE__) && \
    __has_builtin(__builtin_amdgcn_global_load_async_to_lds_b128) && \
    __has_builtin(__builtin_amdgcn_global_store_async_from_lds_b128)
#define USE_ASYNC_LDS 1
#else
#define USE_ASYNC_LDS 0
#endif

#if __has_builtin(__builtin_amdgcn_s_wait_asynccnt)
#define ASYNC_WAIT0() __builtin_amdgcn_s_wait_asynccnt(0)
#else
#define ASYNC_WAIT0() asm volatile("s_wait_asynccnt 0" ::: "memory")
#endif

// ---------------------------------------------------------------------------
// Kernel: token-parallel expansion (defined FIRST so its asm shows in the
// disasm snippet). One wave32 per token: the 2KB row is DMA'd once into LDS
// (4 x global_load_async_to_lds_b128) and streamed to dur consecutive output
// frames with global_store_async_from_lds_b128, fenced by s_wait_asynccnt.
// ---------------------------------------------------------------------------
__global__ void __launch_bounds__(256)
lr_expand(const float* __restrict__ x, const int* __restrict__ cum,
          float* __restrict__ out, int max_len) {
  const int lane  = threadIdx.x & 31;
  const int wid   = threadIdx.x >> 5;           // 8 waves per block
  const int token = blockIdx.x * 8 + wid;
  const int b     = blockIdx.y;

  const int c1   = cum[b * NTK + token];
  const int c0   = (token == 0) ? 0 : cum[b * NTK + token - 1];
  const int dcnt = c1 - c0;
  if (dcnt <= 0) return;

  const vfloat4* __restrict__ src =
      (const vfloat4*)(x + ((size_t)b * NTK + token) * ND);

#if USE_ASYNC_LDS
  __shared__ float tile[8][ND];                 // 2KB per wave, private
  float* lrow = tile[wid];
#pragma unroll
  for (int c = 0; c < 4; ++c) {
    __builtin_amdgcn_global_load_async_to_lds_b128(
        (v4i*)(src + (lane + c * 32)),
        (v4i_lds*)(lrow + (size_t)(lane + c * 32) * 4), 0, 0);
  }
  ASYNC_WAIT0();                                // LDS side of async is OOO
  for (int k = 0; k < dcnt; ++k) {
    vfloat4* dst = (vfloat4*)(out + ((size_t)b * max_len + (c0 + k)) * ND);
#pragma unroll
    for (int c = 0; c < 4; ++c) {
      __builtin_amdgcn_global_store_async_from_lds_b128(
          (v4i*)(dst + (lane + c * 32)),
          (v4i_lds*)(lrow + (size_t)(lane + c * 32) * 4), 0, 0);
    }
  }
  ASYNC_WAIT0();
#else
  // VGPR staging fallback: 16 floats/lane = 4 x global_load_b128
  vfloat4 r0 = src[lane];
  vfloat4 r1 = src[lane + 32];
  vfloat4 r2 = src[lane + 64];
  vfloat4 r3 = src[lane + 96];
  for (int k = 0; k < dcnt; ++k) {
    vfloat4* dst = (vfloat4*)(out + ((size_t)b * max_len + (c0 + k)) * ND);
    __builtin_nontemporal_store(r0, dst + lane);
    __builtin_nontemporal_store(r1, dst + lane + 32);
    __builtin_nontemporal_store(r2, dst + lane + 64);
    __builtin_nontemporal_store(r3, dst + lane + 96);
  }
#endif
}

// ---------------------------------------------------------------------------
// Kernel: per-batch inclusive cumsum via wave32 shuffle scan (5 shfl steps),
// 16 wave sums combined through LDS with a single barrier. Emits cum[B][T],
// mel_len[B], and mel_lens (float) into the d_out tail.
// ---------------------------------------------------------------------------
__global__ void __launch_bounds__(NTK)
lr_cumsum(const int* __restrict__ dur, int* __restrict__ cum,
          int* __restrict__ mel, float* __restrict__ mel_out) {
  const int b    = blockIdx.x;
  const int t    = threadIdx.x;
  const int lane = t & 31;
  const int w    = t >> 5;                      // 16 waves

  int d = dur[b * NTK + t];
  if (d < 0) d = 0;

  // Intra-wave inclusive scan (wave32): 5 shuffle-up steps.
  int v = d;
#pragma unroll
  for (int off = 1; off < 32; off <<= 1) {
    int u = __shfl_up(v, off, 32);
    if (lane >= off) v += u;
  }

  __shared__ int wsum[NWAVES];
  if (lane == 31) wsum[w] = v;
  __syncthreads();

  // Uniform (broadcast) accumulation of preceding wave sums.
  int base = 0;
#pragma unroll
  for (int i = 0; i < NWAVES; ++i) {
    int s = wsum[i];
    base += (i < w) ? s : 0;
  }

  const int c = v + base;
  cum[b * NTK + t] = c;
  if (t == NTK - 1) {
    mel[b] = c;
    mel_out[b] = (float)c;   // mel_lens appended to output, as float
  }
}

// ---------------------------------------------------------------------------
// Kernel: zero-fill only frames j in [mel_len[b], max_len).
// ---------------------------------------------------------------------------
__global__ void __launch_bounds__(256)
lr_pad(const int* __restrict__ mel, float* __restrict__ out, int max_len) {
  const int b  = blockIdx.y;
  const int ml = mel[b];
  const size_t i   = (size_t)blockIdx.x * 256 + threadIdx.x;  // float4 index
  const size_t nf4 = (size_t)max_len * ND4;
  if (i >= nf4) return;
  const int j = (int)(i >> 7);                  // i / ND4 (ND4 == 128)
  if (j < ml) return;                           // valid region: no traffic
  vfloat4 z = (vfloat4)0.0f;
  __builtin_nontemporal_store(z, (vfloat4*)(out + (size_t)b * max_len * ND) + i);
}

// ---------------------------------------------------------------------------
extern "C" void kernel_launch(void* const* d_in, const int* in_sizes, int n_in,
                              void* d_out, int out_size, void* d_ws, size_t ws_size,
                              hipStream_t stream) {
  (void)in_sizes; (void)n_in; (void)ws_size;
  const float* x   = (const float*)d_in[0];
  const int*   dur = (const int*)d_in[1];
  float*       out = (float*)d_out;

  int* cum = (int*)d_ws;          // NB*NTK ints
  int* mel = cum + NB * NTK;      // NB ints

  // out_size = B*max_len*D + B  (out tensor + mel_lens tail)
  int max_len = (out_size - NB) / (NB * ND);
  if (max_len < 1) max_len = 1;
  float* mel_out = out + (size_t)NB * max_len * ND;

  lr_cumsum<<<dim3(NB), dim3(NTK), 0, stream>>>(dur, cum, mel, mel_out);
  lr_expand<<<dim3(NTK / 8, NB), dim3(256), 0, stream>>>(x, cum, out, max_len);
  const int nf4 = max_len * ND4;
  lr_pad<<<dim3((nf4 + 255) / 256, NB), dim3(256), 0, stream>>>(mel, out, max_len);
}